// MambaEncoder_77206332113652
// MI455X (gfx1250) — compile-verified
//
#include <hip/hip_runtime.h>
#include <hip/hip_bf16.h>
#include <math.h>

// ---------------- model constants ----------------
#define E_LAYERS 4
#define D_MODEL  512
#define D_STATE  16
#define D_CONV   4
#define D_INNER  1024          // EXPAND * D_MODEL
#define DT_RANK  32
#define BATCH    4
#define SEQ      1024
#define NTOK     (BATCH * SEQ)     // 4096
#define XPROJ_N  (DT_RANK + 2 * D_STATE)  // 64
#define EPS      1e-5f

typedef float v2f __attribute__((ext_vector_type(2)));
typedef float v8f __attribute__((ext_vector_type(8)));

// =====================================================================
// LayerNorm: one block (256 threads) per row of 512
// =====================================================================
__global__ __launch_bounds__(256) void ln_kernel(const float* __restrict__ x,
                                                 const float* __restrict__ w,
                                                 const float* __restrict__ b,
                                                 float* __restrict__ out) {
    const int row = blockIdx.x;
    const int t   = threadIdx.x;
    const float* xr = x + (size_t)row * D_MODEL;
    __shared__ float red[256];

    float a0 = xr[t];
    float a1 = xr[t + 256];
    red[t] = a0 + a1;
    __syncthreads();
    #pragma unroll
    for (int off = 128; off > 0; off >>= 1) {
        if (t < off) red[t] += red[t + off];
        __syncthreads();
    }
    const float mu = red[0] * (1.0f / D_MODEL);
    __syncthreads();

    const float d0 = a0 - mu, d1 = a1 - mu;
    red[t] = d0 * d0 + d1 * d1;
    __syncthreads();
    #pragma unroll
    for (int off = 128; off > 0; off >>= 1) {
        if (t < off) red[t] += red[t + off];
        __syncthreads();
    }
    const float rs = rsqrtf(red[0] * (1.0f / D_MODEL) + EPS);

    float* orow = out + (size_t)row * D_MODEL;
    orow[t]       = d0 * rs * w[t]       + b[t];
    orow[t + 256] = d1 * rs * w[t + 256] + b[t + 256];
}

// =====================================================================
// async global -> LDS copy of one 16-byte chunk per lane (gfx1250,
// tracked by ASYNCcnt).  Inline asm is portable across both toolchains'
// differing builtin arities.  LDS byte address = low 32 bits of the
// generic pointer (flat->LDS truncation rule).
// =====================================================================
__device__ __forceinline__ void async_copy_b128(void* lds_dst, const float* gsrc) {
    const uint32_t lds_off = (uint32_t)(uintptr_t)lds_dst;
    const uint64_t gaddr   = (uint64_t)(uintptr_t)gsrc;
    asm volatile("global_load_async_to_lds_b128 %0, %1, off"
                 :: "v"(lds_off), "v"(gaddr)
                 : "memory");
}

// =====================================================================
// Generic NT WMMA GEMM:  C[M,N] = act(A[M,K] @ W[N,K]^T + bias) + resid
//   block = 256 threads (8 waves), block tile 128(M) x 64(N), Ktile 32
//   A and W tiles double-buffered in LDS via async-to-LDS loads,
//   v_wmma_f32_16x16x4_f32 compute.  act: 0 = none, 1 = softplus
// =====================================================================
__global__ __launch_bounds__(256) void wmma_gemm(const float* __restrict__ A, int lda,
                                                 const float* __restrict__ W,
                                                 const float* __restrict__ bias,
                                                 const float* __restrict__ resid,
                                                 float* __restrict__ C,
                                                 int M, int N, int K, int act) {
    // padded row stride 36 floats = 144 B (multiple of 16 B for b128 async writes)
    __shared__ __align__(16) float As[2][128][36];
    __shared__ __align__(16) float Bs[2][64][36];

    const int tid  = threadIdx.x;
    const int wave = tid >> 5;
    const int lane = tid & 31;
    const int nl   = lane & 15;     // M index (A frag) / N index (B,C frags)
    const int kh   = lane >> 4;     // K-half (A/B frags) / M-half (C frag)

    const int mBase = blockIdx.y * 128;
    const int m0    = mBase + wave * 16;
    const int n0    = blockIdx.x * 64;

    // cooperative-load slot decode (float4 granularity)
    const int rowA0 = tid >> 3;              // A tile: 1024 slots, 4 per thread
    const int c4    = (tid & 7) << 2;

    v8f acc[4];
    #pragma unroll
    for (int t = 0; t < 4; ++t) acc[t] = (v8f){0.f,0.f,0.f,0.f,0.f,0.f,0.f,0.f};

    const int nTiles = K >> 5;

    // ---- issue async loads for one (A,B) K-tile into buffer `buf` ----
    auto issue_tile = [&](int buf, int k0) {
        #pragma unroll
        for (int e = 0; e < 4; ++e) {            // A: 128 rows x 8 float4
            const int row = rowA0 + e * 32;
            async_copy_b128(&As[buf][row][c4],
                            A + (size_t)(mBase + row) * lda + k0 + c4);
        }
        #pragma unroll
        for (int e = 0; e < 2; ++e) {            // W: 64 rows x 8 float4
            const int row = rowA0 + e * 32;
            async_copy_b128(&Bs[buf][row & 63][c4],
                            W + (size_t)(n0 + (row & 63)) * K + k0 + c4);
        }
    };

    issue_tile(0, 0);

    for (int t = 0; t < nTiles; ++t) {
        const int buf = t & 1;
        if (t > 0) __syncthreads();              // prior compute done before overwrite
        if (t + 1 < nTiles) {
            issue_tile(buf ^ 1, (t + 1) << 5);   // prefetch next tile (6 async ops)
            asm volatile("s_wait_asynccnt 0x6" ::: "memory");  // tile t landed
        } else {
            asm volatile("s_wait_asynccnt 0x0" ::: "memory");
        }
        __syncthreads();                         // tile t visible to all waves

        #pragma unroll
        for (int kk = 0; kk < 32; kk += 4) {
            // A fragment: lanes 0-15 -> M=nl,K=kk..kk+1 ; lanes 16-31 -> K=kk+2..kk+3
            const v2f a = *(const v2f*)(&As[buf][wave * 16 + nl][kk + 2 * kh]);
            #pragma unroll
            for (int u = 0; u < 4; ++u) {
                // B fragment: Bmat[k][n] = W[n][k]
                const v2f b = *(const v2f*)(&Bs[buf][u * 16 + nl][kk + 2 * kh]);
                acc[u] = __builtin_amdgcn_wmma_f32_16x16x4_f32(
                    false, a, false, b, (short)0, acc[u], false, false);
            }
        }
    }

    // epilogue: C layout — VGPR r: lanes0-15 M=r, lanes16-31 M=r+8; N = lane&15
    #pragma unroll
    for (int t = 0; t < 4; ++t) {
        const int n = n0 + t * 16 + nl;
        const float bval = bias ? bias[n] : 0.0f;
        #pragma unroll
        for (int r = 0; r < 8; ++r) {
            const int m = m0 + r + 8 * kh;
            float v = acc[t][r] + bval;
            if (act == 1) {                         // softplus
                v = (v > 20.0f) ? v : log1pf(__expf(v));
            }
            if (resid) v += resid[(size_t)m * N + n];
            C[(size_t)m * N + n] = v;
        }
    }
}

// =====================================================================
// Causal depthwise conv (k=4) + bias + SiLU, reading the xh half of xz
// one thread per (b, l, d)
// =====================================================================
__global__ __launch_bounds__(256) void conv_silu_kernel(const float* __restrict__ xz,
                                                        const float* __restrict__ cw,
                                                        const float* __restrict__ cb,
                                                        float* __restrict__ xh) {
    const long idx = (long)blockIdx.x * 256 + threadIdx.x;  // B*L*D_INNER
    const int d = (int)(idx & (D_INNER - 1));
    const int l = (int)((idx >> 10) & (SEQ - 1));
    const int b = (int)(idx >> 20);

    const float* w = cw + (size_t)d * D_CONV;
    float acc = cb[d];
    #pragma unroll
    for (int j = 0; j < D_CONV; ++j) {
        const int ls = l - (D_CONV - 1) + j;
        if (ls >= 0) acc += w[j] * xz[((size_t)b * SEQ + ls) * (2 * D_INNER) + d];
    }
    // SiLU
    acc = acc / (1.0f + __expf(-acc));
    xh[((size_t)b * SEQ + l) * D_INNER + d] = acc;
}

// =====================================================================
// Selective scan, fused with u*D and SiLU(z) gating.
// one thread per (b, d); 16-wide state in registers; 1024 serial steps
// =====================================================================
__global__ __launch_bounds__(256) void scan_kernel(const float* __restrict__ xh,
                                                   const float* __restrict__ delta,
                                                   const float* __restrict__ xdbc,
                                                   const float* __restrict__ xz,
                                                   const float* __restrict__ A_log,
                                                   const float* __restrict__ Dv,
                                                   float* __restrict__ yg) {
    const int gid = blockIdx.x * 256 + threadIdx.x;   // B * D_INNER
    const int b = gid >> 10;
    const int d = gid & (D_INNER - 1);

    float A[D_STATE], h[D_STATE];
    #pragma unroll
    for (int n = 0; n < D_STATE; ++n) {
        A[n] = -__expf(A_log[(size_t)d * D_STATE + n]);
        h[n] = 0.0f;
    }
    const float Dd = Dv[d];

    for (int l = 0; l < SEQ; ++l) {
        const size_t rb = (size_t)b * SEQ + l;
        const float dt = delta[rb * D_INNER + d];
        const float u  = xh[rb * D_INNER + d];
        const float du = dt * u;
        const float* bc = xdbc + rb * XPROJ_N;
        float y = 0.0f;
        #pragma unroll
        for (int n = 0; n < D_STATE; ++n) {
            const float dA = __expf(dt * A[n]);
            h[n] = dA * h[n] + du * bc[DT_RANK + n];
            y += h[n] * bc[DT_RANK + D_STATE + n];
        }
        y += u * Dd;
        const float z = xz[rb * (2 * D_INNER) + D_INNER + d];
        y *= z / (1.0f + __expf(-z));   // * SiLU(z)
        yg[rb * D_INNER + d] = y;
    }
}

// =====================================================================
// host-side launcher
// =====================================================================
extern "C" void kernel_launch(void* const* d_in, const int* in_sizes, int n_in,
                              void* d_out, int out_size, void* d_ws, size_t ws_size,
                              hipStream_t stream) {
    const float* x_in      = (const float*)d_in[0];
    const float* norm_w    = (const float*)d_in[1];
    const float* norm_b    = (const float*)d_in[2];
    const float* in_proj_w = (const float*)d_in[3];
    const float* conv_w    = (const float*)d_in[4];
    const float* conv_b    = (const float*)d_in[5];
    const float* x_proj_w  = (const float*)d_in[6];
    const float* dt_proj_w = (const float*)d_in[7];
    const float* dt_proj_b = (const float*)d_in[8];
    const float* A_log     = (const float*)d_in[9];
    const float* Dvec      = (const float*)d_in[10];
    const float* out_proj_w= (const float*)d_in[11];

    // workspace layout (floats)
    float* ws = (float*)d_ws;
    const size_t SZ_NORMED = (size_t)NTOK * D_MODEL;        // 2M
    const size_t SZ_XZ     = (size_t)NTOK * 2 * D_INNER;    // 8M
    const size_t SZ_XH     = (size_t)NTOK * D_INNER;        // 4M
    const size_t SZ_XDBC   = (size_t)NTOK * XPROJ_N;        // 256K
    const size_t SZ_DELTA  = (size_t)NTOK * D_INNER;        // 4M
    float* normed = ws;
    float* xz     = normed + SZ_NORMED;
    float* xh     = xz     + SZ_XZ;
    float* xdbc   = xh     + SZ_XH;
    float* delta  = xdbc   + SZ_XDBC;
    float* yg     = delta  + SZ_DELTA;

    const float* xcur = x_in;
    for (int l = 0; l < E_LAYERS; ++l) {
        // 1) LayerNorm
        ln_kernel<<<NTOK, 256, 0, stream>>>(
            xcur, norm_w + (size_t)l * D_MODEL, norm_b + (size_t)l * D_MODEL, normed);

        // 2) in_proj: [4096,512] x [2048,512]^T -> xz [4096,2048]
        wmma_gemm<<<dim3((2 * D_INNER) / 64, NTOK / 128), 256, 0, stream>>>(
            normed, D_MODEL, in_proj_w + (size_t)l * 2 * D_INNER * D_MODEL,
            nullptr, nullptr, xz, NTOK, 2 * D_INNER, D_MODEL, 0);

        // 3) causal dw-conv + SiLU on the xh half
        conv_silu_kernel<<<(NTOK * D_INNER) / 256, 256, 0, stream>>>(
            xz, conv_w + (size_t)l * D_INNER * D_CONV, conv_b + (size_t)l * D_INNER, xh);

        // 4) x_proj: [4096,1024] x [64,1024]^T -> xdbc [4096,64]
        wmma_gemm<<<dim3(XPROJ_N / 64, NTOK / 128), 256, 0, stream>>>(
            xh, D_INNER, x_proj_w + (size_t)l * XPROJ_N * D_INNER,
            nullptr, nullptr, xdbc, NTOK, XPROJ_N, D_INNER, 0);

        // 5) dt_proj + bias + softplus: [4096,32] x [1024,32]^T -> delta [4096,1024]
        wmma_gemm<<<dim3(D_INNER / 64, NTOK / 128), 256, 0, stream>>>(
            xdbc, XPROJ_N, dt_proj_w + (size_t)l * D_INNER * DT_RANK,
            dt_proj_b + (size_t)l * D_INNER, nullptr, delta, NTOK, D_INNER, DT_RANK, 1);

        // 6) selective scan + u*D + SiLU(z) gate -> yg [4096,1024]
        scan_kernel<<<(BATCH * D_INNER) / 256, 256, 0, stream>>>(
            xh, delta, xdbc, xz,
            A_log + (size_t)l * D_INNER * D_STATE, Dvec + (size_t)l * D_INNER, yg);

        // 7) out_proj + residual, writing directly into d_out layer slice
        float* outl = (float*)d_out + (size_t)l * NTOK * D_MODEL;
        wmma_gemm<<<dim3(D_MODEL / 64, NTOK / 128), 256, 0, stream>>>(
            yg, D_INNER, out_proj_w + (size_t)l * D_MODEL * D_INNER,
            nullptr, xcur, outl, NTOK, D_MODEL, D_INNER, 0);

        xcur = outl;   // residual stream for next layer
    }
}